// ForceModule_49194555408957
// MI455X (gfx1250) — compile-verified
//
#include <hip/hip_runtime.h>
#include <math.h>

// ---------------------------------------------------------------------------
// Stage 2 (dominant): sum_g h_g * exp(-0.5*||(s-c_g)/w_g||^2) over 8,388,608
// gaussians. 160 MB of reads -> HBM-bound, ~6.9 us floor at 23.3 TB/s
// (working set < 192 MB L2, so keep RT caching for graph-replay L2 hits).
// Stage 1 (tiny): 45-feature MLP; 128x128 layer done with V_WMMA_F32_16X16X4_F32
// (full fp32 precision WMMA) -- one N-tile per wave, 8 waves, 32 k-steps.
// ---------------------------------------------------------------------------

typedef __attribute__((ext_vector_type(2))) float v2f;
typedef __attribute__((ext_vector_type(8))) float v8f;

#define NPAIRS 45
#define HDIM   128

__device__ __constant__ int c_pi[NPAIRS] = {
    1,1,1,1,1,1,1,1,1,
    4,4,4,4,4,4,4,4,
    5,5,5,5,5,5,5,
    6,6,6,6,6,6,
    8,8,8,8,8,
    10,10,10,10,
    14,14,14,
    15,15,
    16};
__device__ __constant__ int c_pj[NPAIRS] = {
    4,5,6,8,10,14,15,16,18,
    5,6,8,10,14,15,16,18,
    6,8,10,14,15,16,18,
    8,10,14,15,16,18,
    10,14,15,16,18,
    14,15,16,18,
    15,16,18,
    16,18,
    18};

// ---------------------------------------------------------------------------
// Kernel 1: featurize + 3-layer MLP. One block of 256 threads (8 wave32s).
// Writes s[0..1] into ws.
// ---------------------------------------------------------------------------
__global__ __launch_bounds__(256) void mlp_kernel(
    const float* __restrict__ pos,
    const float* __restrict__ W1, const float* __restrict__ b1,
    const float* __restrict__ W2, const float* __restrict__ b2,
    const float* __restrict__ W3, const float* __restrict__ b3,
    float* __restrict__ s_out)
{
    __shared__ float distS[48];
    __shared__ float h1s[HDIM];
    __shared__ float h2s[HDIM];

    const int tid = threadIdx.x;

    // --- featurize: 45 pairwise distances ---
    if (tid < NPAIRS) {
        const int i = c_pi[tid], j = c_pj[tid];
        const float dx = pos[j * 3 + 0] - pos[i * 3 + 0];
        const float dy = pos[j * 3 + 1] - pos[i * 3 + 1];
        const float dz = pos[j * 3 + 2] - pos[i * 3 + 2];
        distS[tid] = sqrtf(dx * dx + dy * dy + dz * dz);
    }
    __syncthreads();

    // --- layer 1: h1 = tanh(x @ W1 + b1), K=45 (scalar; K not WMMA-shaped) ---
    if (tid < HDIM) {
        float acc = b1[tid];
#pragma unroll
        for (int k = 0; k < NPAIRS; ++k)
            acc += distS[k] * W1[k * HDIM + tid];   // coalesced across tid
        h1s[tid] = tanhf(acc);
    }
    __syncthreads();

    // --- layer 2: h2 = tanh(h1 @ W2 + b2) via V_WMMA_F32_16X16X4_F32 ---
    // Wave w owns output columns [16w, 16w+16). M=16 with only row 0 live.
    // A 16x4 f32 frag: V0 = {K=0 | K=2}, V1 = {K=1 | K=3} (lanes 0-15 | 16-31).
    // B 4x16  f32 frag: same K split, N = lane & 15.
    // A is built with an unconditional LDS broadcast load times a 0/1 lane
    // mask -> no EXEC-divergent ds_loads, EXEC stays all-1 through the loop.
    const int wave = tid >> 5;
    const int lane = tid & 31;
    const int nl   = lane & 15;
    const int n    = wave * 16 + nl;        // output column
    const int kk   = (lane >= 16) ? 2 : 0;  // K half handled by this lane group
    const float am = (nl == 0) ? 1.0f : 0.0f;  // A row M=0 only

    v8f acc = {};  // C = 0
#pragma unroll 4
    for (int k0 = 0; k0 < HDIM; k0 += 4) {
        v2f a, b;
        a[0] = am * h1s[k0 + kk + 0];
        a[1] = am * h1s[k0 + kk + 1];
        b[0] = W2[(k0 + kk + 0) * HDIM + n];          // row-major W2[k][n]
        b[1] = W2[(k0 + kk + 1) * HDIM + n];
        acc = __builtin_amdgcn_wmma_f32_16x16x4_f32(
            /*neg_a=*/false, a, /*neg_b=*/false, b,
            /*c_mod=*/(short)0, acc, /*reuse_a=*/false, /*reuse_b=*/false);
    }
    // D row M=0 lives in acc[0] on lanes 0-15.
    if (lane < 16)
        h2s[n] = tanhf(acc[0] + b2[n]);
    __syncthreads();

    // --- layer 3: s = h2 @ W3 + b3 (128x2, scalar) ---
    if (tid < 2) {
        float s = b3[tid];
#pragma unroll 8
        for (int k = 0; k < HDIM; ++k)
            s += h2s[k] * W3[k * 2 + tid];
        s_out[tid] = s;
    }
}

// ---------------------------------------------------------------------------
// Kernel 2: streaming gaussian reduction. 4 gaussians/thread/iter via b128
// loads; exactly uniform trip count (no divergence at the wave reduce).
// Fast-path div/exp (~1e-6 rel err) keeps VALU well under the HBM floor.
// Writes one partial per block (deterministic two-pass reduction, no atomics).
// ---------------------------------------------------------------------------
__global__ __launch_bounds__(256) void bias_kernel(
    const float* __restrict__ heights,
    const float* __restrict__ centers,
    const float* __restrict__ widths,
    const float* __restrict__ s_in,
    float* __restrict__ partials,
    int nGroups)   // nGroups = G/4
{
    const float s0 = s_in[0];
    const float s1 = s_in[1];

    const float4* __restrict__ h4 = (const float4*)heights;
    const float4* __restrict__ c4 = (const float4*)centers;
    const float4* __restrict__ w4 = (const float4*)widths;

    float sum = 0.0f;
    const int stride = gridDim.x * blockDim.x;
    for (int i = blockIdx.x * blockDim.x + threadIdx.x; i < nGroups; i += stride) {
        if (i + stride < nGroups) {             // prefetch next grid-stride tile
            __builtin_prefetch(&h4[i + stride], 0, 0);
            __builtin_prefetch(&c4[2 * (i + stride)], 0, 0);
            __builtin_prefetch(&w4[2 * (i + stride)], 0, 0);
        }
        const float4 h  = h4[i];
        const float4 cA = c4[2 * i + 0];
        const float4 cB = c4[2 * i + 1];
        const float4 wA = w4[2 * i + 0];
        const float4 wB = w4[2 * i + 1];

        float dx, dy;
        dx = __fdividef(s0 - cA.x, wA.x); dy = __fdividef(s1 - cA.y, wA.y);
        sum += h.x * __expf(-0.5f * (dx * dx + dy * dy));
        dx = __fdividef(s0 - cA.z, wA.z); dy = __fdividef(s1 - cA.w, wA.w);
        sum += h.y * __expf(-0.5f * (dx * dx + dy * dy));
        dx = __fdividef(s0 - cB.x, wB.x); dy = __fdividef(s1 - cB.y, wB.y);
        sum += h.z * __expf(-0.5f * (dx * dx + dy * dy));
        dx = __fdividef(s0 - cB.z, wB.z); dy = __fdividef(s1 - cB.w, wB.w);
        sum += h.w * __expf(-0.5f * (dx * dx + dy * dy));
    }

    // wave32 tree reduce (all lanes active: trip count is uniform)
    for (int m = 16; m > 0; m >>= 1) sum += __shfl_down(sum, m, 32);

    __shared__ float wsum[8];
    if ((threadIdx.x & 31) == 0) wsum[threadIdx.x >> 5] = sum;
    __syncthreads();
    if (threadIdx.x < 32) {
        float v = (threadIdx.x < 8) ? wsum[threadIdx.x] : 0.0f;
        for (int m = 4; m > 0; m >>= 1) v += __shfl_down(v, m, 32);
        if (threadIdx.x == 0) partials[blockIdx.x] = v;
    }
}

// ---------------------------------------------------------------------------
// Kernel 3: deterministic final reduce of per-block partials -> out[0].
// ---------------------------------------------------------------------------
__global__ __launch_bounds__(256) void final_reduce(
    const float* __restrict__ partials, int n, float* __restrict__ out)
{
    float sum = 0.0f;
    for (int i = threadIdx.x; i < n; i += 256) sum += partials[i];
    for (int m = 16; m > 0; m >>= 1) sum += __shfl_down(sum, m, 32);

    __shared__ float wsum[8];
    if ((threadIdx.x & 31) == 0) wsum[threadIdx.x >> 5] = sum;
    __syncthreads();
    if (threadIdx.x < 32) {
        float v = (threadIdx.x < 8) ? wsum[threadIdx.x] : 0.0f;
        for (int m = 4; m > 0; m >>= 1) v += __shfl_down(v, m, 32);
        if (threadIdx.x == 0) out[0] = v;
    }
}

extern "C" void kernel_launch(void* const* d_in, const int* in_sizes, int n_in,
                              void* d_out, int out_size, void* d_ws, size_t ws_size,
                              hipStream_t stream) {
    const float* pos     = (const float*)d_in[0];
    const float* heights = (const float*)d_in[1];
    const float* centers = (const float*)d_in[2];
    const float* widths  = (const float*)d_in[3];
    const float* W1      = (const float*)d_in[4];
    const float* b1      = (const float*)d_in[5];
    const float* W2      = (const float*)d_in[6];
    const float* b2      = (const float*)d_in[7];
    const float* W3      = (const float*)d_in[8];
    const float* b3      = (const float*)d_in[9];
    float* out = (float*)d_out;
    float* ws  = (float*)d_ws;      // ws[0..1] = s, ws[2..2+BLOCKS) = partials

    const int G       = in_sizes[1];
    const int nGroups = G / 4;      // 4 gaussians per b128-load group
    const int BLOCKS  = 2048;       // 16384 wave32s in flight -> saturate HBM

    mlp_kernel<<<1, 256, 0, stream>>>(pos, W1, b1, W2, b2, W3, b3, ws);
    bias_kernel<<<BLOCKS, 256, 0, stream>>>(heights, centers, widths,
                                            ws, ws + 2, nGroups);
    final_reduce<<<1, 256, 0, stream>>>(ws + 2, BLOCKS, out);
}